// EfficientScaledDotProductAttention_2920577761328
// MI455X (gfx1250) — compile-verified
//
#include <hip/hip_runtime.h>
#include <hip/hip_bf16.h>

// out[b] = V[b] @ (K[b]^T @ Q[b])
//   B=16, S=2048, D=64, all f32.
// Memory-bound (AI ~16 FLOP/B); stream inputs once, use V_WMMA_F32_16X16X4_F32.

typedef __attribute__((ext_vector_type(2))) float v2f;
typedef __attribute__((ext_vector_type(8))) float v8f;

#define SEQ 2048
#define DH  64
#define NB  16

// ---------------------------------------------------------------------------
// Phase 1: M[b] (64x64) = K[b]^T @ Q[b], reduction over SEQ.
// grid = (16 tiles, NB batches), block = 256 (8 waves).
// Each wave owns one 16x16 tile of M over a 256-row slice of S; partials
// reduced across waves via LDS (no FP atomics -> deterministic).
// ---------------------------------------------------------------------------
__global__ __launch_bounds__(256)
void esdpa_ktq_kernel(const float* __restrict__ Q,
                      const float* __restrict__ K,
                      float* __restrict__ M) {
  __shared__ float red[8 * 256];          // 8 waves x 16x16 partial tile

  const int b    = blockIdx.y;
  const int tile = blockIdx.x;            // 0..15
  const int t1   = tile >> 2;             // row tile of M (d' index)
  const int t2   = tile & 3;              // col tile of M (d index)

  const int wave = threadIdx.x >> 5;
  const int lane = threadIdx.x & 31;
  const int half = lane >> 4;             // 0: K=0,1   1: K=2,3 (A/B layout)
  const int l    = lane & 15;

  const float* __restrict__ Kb = K + (size_t)b * SEQ * DH;
  const float* __restrict__ Qb = Q + (size_t)b * SEQ * DH;

  const int d1 = t1 * 16 + l;             // A column (row of M)
  const int d2 = t2 * 16 + l;             // B column (col of M)

  v8f acc = {};
  int k0 = wave * (SEQ / 8);
  const int kend = k0 + (SEQ / 8);
  for (; k0 < kend; k0 += 4) {
    const int kr = k0 + 2 * half;
    v2f a, bb;
    // A[m=d1, kk] = K[k0+kk, d1]  (K^T): lanes 0-15 -> kk=0,1 ; 16-31 -> kk=2,3
    a.x  = Kb[(size_t)kr       * DH + d1];
    a.y  = Kb[(size_t)(kr + 1) * DH + d1];
    // B[kk, n=d2] = Q[k0+kk, d2]
    bb.x = Qb[(size_t)kr       * DH + d2];
    bb.y = Qb[(size_t)(kr + 1) * DH + d2];
    acc = __builtin_amdgcn_wmma_f32_16x16x4_f32(
        /*neg_a=*/false, a, /*neg_b=*/false, bb,
        /*c_mod=*/(short)0, acc, /*reuse_a=*/false, /*reuse_b=*/false);
  }

  // Spill per-wave 16x16 partial (C layout: VGPR r = row r+8*half, col = l).
#pragma unroll
  for (int r = 0; r < 8; ++r)
    red[wave * 256 + (r + 8 * half) * 16 + l] = acc[r];
  __syncthreads();

  // Tree-free reduction: thread t sums element t across the 8 wave partials.
  float s = 0.f;
#pragma unroll
  for (int w = 0; w < 8; ++w) s += red[w * 256 + threadIdx.x];

  const int row = threadIdx.x >> 4;
  const int col = threadIdx.x & 15;
  M[(size_t)b * DH * DH + (size_t)(t1 * 16 + row) * DH + (t2 * 16 + col)] = s;
}

// ---------------------------------------------------------------------------
// Phase 2: out[b] (2048x64) = V[b] @ M[b].
// grid = (16, NB), block = 256 (8 waves). M staged once per block into LDS;
// each wave produces a 16x64 output tile (4 accumulators, 16 WMMAs each).
// ---------------------------------------------------------------------------
__global__ __launch_bounds__(256)
void esdpa_vm_kernel(const float* __restrict__ V,
                     const float* __restrict__ M,
                     float* __restrict__ Out) {
  __shared__ float ldsM[DH * DH];         // 16 KB

  const int b = blockIdx.y;
  for (int i = threadIdx.x; i < DH * DH; i += 256)
    ldsM[i] = M[(size_t)b * DH * DH + i];
  __syncthreads();

  const int wave = threadIdx.x >> 5;
  const int lane = threadIdx.x & 31;
  const int half = lane >> 4;
  const int l    = lane & 15;
  const int row0 = (blockIdx.x * 8 + wave) * 16;   // V row tile

  const float* __restrict__ Vb = V + (size_t)b * SEQ * DH;
  float* __restrict__ Ob       = Out + (size_t)b * SEQ * DH;

#pragma unroll
  for (int t2 = 0; t2 < 4; ++t2) {
    v8f acc = {};
#pragma unroll
    for (int dp = 0; dp < DH; dp += 4) {
      // A[m, kk] = V[row0+m, dp+kk]: per-lane contiguous float2 load.
      v2f a, bb;
      const float* ap = &Vb[(size_t)(row0 + l) * DH + dp + 2 * half];
      a.x = ap[0];
      a.y = ap[1];
      // B[kk, n] = M[dp+kk, t2*16+n] from LDS.
      const int br = dp + 2 * half;
      bb.x = ldsM[br       * DH + t2 * 16 + l];
      bb.y = ldsM[(br + 1) * DH + t2 * 16 + l];
      acc = __builtin_amdgcn_wmma_f32_16x16x4_f32(
          false, a, false, bb, (short)0, acc, false, false);
    }
#pragma unroll
    for (int r = 0; r < 8; ++r)
      Ob[(size_t)(row0 + r + 8 * half) * DH + (t2 * 16 + l)] = acc[r];
  }
}

extern "C" void kernel_launch(void* const* d_in, const int* in_sizes, int n_in,
                              void* d_out, int out_size, void* d_ws, size_t ws_size,
                              hipStream_t stream) {
  const float* Q = (const float*)d_in[0];   // query  [16,2048,64]
  const float* K = (const float*)d_in[1];   // key    [16,2048,64]
  const float* V = (const float*)d_in[2];   // value  [16,2048,64]
  float* out = (float*)d_out;               // [16,2048,64]
  float* M   = (float*)d_ws;                // 16*64*64 f32 = 256 KB scratch

  (void)in_sizes; (void)n_in; (void)out_size; (void)ws_size;

  // Phase 1: per-batch 64x64 M = K^T Q  (overwrites scratch, no init needed)
  esdpa_ktq_kernel<<<dim3(16, NB), 256, 0, stream>>>(Q, K, M);
  // Phase 2: out = V @ M  (stream-ordered after phase 1)
  esdpa_vm_kernel<<<dim3(SEQ / 16 / 8, NB), 256, 0, stream>>>(V, M, out);
}